// LinearAttention_77884936946291
// MI455X (gfx1250) — compile-verified
//
#include <hip/hip_runtime.h>
#include <math.h>

// ---------------------------------------------------------------------------
// Linear attention on MI455X (gfx1250, wave32, WMMA + Tensor Data Mover).
//   N=4, S=8192, D=512, F=H*HD=512, OUT=512, f32 end-to-end.
// 4 GEMMs (K,V,Q,O) via V_WMMA_F32_16X16X4_F32 (full f32 precision).
// Weights are pre-interleaved once into k-pair rows (Wt[p][f*2+e]) so a
// single ds_load_b64 yields a complete WMMA B fragment in an adjacent VGPR
// pair. Panels are staged into LDS by TDM (tensor_load_to_lds), double-
// buffered, pipelined against WMMA with s_wait_tensorcnt + barriers.
// ---------------------------------------------------------------------------

typedef __attribute__((ext_vector_type(2))) float v2f;
typedef __attribute__((ext_vector_type(8))) float v8f;
typedef unsigned int u32x4 __attribute__((ext_vector_type(4)));
typedef int i32x8 __attribute__((ext_vector_type(8)));
typedef int i32x4 __attribute__((ext_vector_type(4)));

#define S_LEN   8192
#define D_IN    512
#define F_DIM   512
#define OUT_DIM 512
#define EPS     1e-6f

#define KC        32                   // K depth per chunk (16 pair-rows)
#define XS_STRIDE 516                  // 512 + TDM pad (2 dw per 256 dw)
#define PSTRIDE   1056                 // 1024 + TDM pad (8 dw per 256 dw)
#define WB_PANEL  (16 * PSTRIDE)       // one W panel, floats
#define WT_SZ     (256 * 1024)         // one interleaved weight, floats

#if defined(__has_builtin)
#if __has_builtin(__builtin_amdgcn_tensor_load_to_lds)
#define HAS_TDM 1
#endif
#endif
#ifndef HAS_TDM
#define HAS_TDM 0
#endif

__device__ __forceinline__ float fmapf(float v) {   // elu(v)+1
  return v > 0.f ? v + 1.f : __expf(v);
}
__device__ __forceinline__ float geluf(float v) {   // tanh-approx gelu via exp
  const float k0 = 0.7978845608028654f, k1 = 0.044715f;
  const float z = k0 * (v + k1 * v * v * v);
  const float e = __expf(2.f * z);
  const float t = 1.f - 2.f / (e + 1.f);            // tanh(z)
  return 0.5f * v * (1.f + t);
}

// ---------------------------------------------------------------------------
// Tensor Data Mover: build D# per CDNA5 ISA and issue.
// 6-arg builtin (clang-23): (u32x4 g0, i32x8 g1, i32x4 g2, i32x4 g3,
//                            i32x8 extra, i32 cpol).
// pad: every 256 dwords stored, insert (pad_code+1) dwords into LDS address.
// ---------------------------------------------------------------------------
#if HAS_TDM
__device__ __forceinline__ void tdm_load(
    unsigned int lds_off, unsigned long long gaddr,
    unsigned int td0, unsigned int td1, unsigned int td2,
    unsigned int t0, unsigned int t1, unsigned int t2,
    unsigned long long s0, unsigned long long s1, unsigned int pad_code) {
  u32x4 g0;
  g0[0] = 1u;                                   // count=1, user descriptor
  g0[1] = lds_off;                              // lds_addr (bytes)
  g0[2] = (unsigned int)gaddr;                  // global_addr[31:0]
  g0[3] = (unsigned int)((gaddr >> 32) & 0x01FFFFFFu) | (2u << 30);  // type=2
  i32x8 g1;
  g1[0] = (int)((2u << 16) | (1u << 20) | (7u << 22) | (pad_code << 25));
  g1[1] = (int)((td0 & 0xFFFFu) << 16);                       // abar=0|td0.lo
  g1[2] = (int)((td0 >> 16) | ((td1 & 0xFFFFu) << 16));
  g1[3] = (int)((td1 >> 16) | (t0 << 16));
  g1[4] = (int)(t1 | (t2 << 16));
  g1[5] = (int)(unsigned int)(s0 & 0xFFFFFFFFull);
  g1[6] = (int)((unsigned int)(s0 >> 32) | ((unsigned int)(s1 & 0xFFFFull) << 16));
  g1[7] = (int)(unsigned int)(s1 >> 16);
  i32x4 g2; g2[0] = (int)td2; g2[1] = 0; g2[2] = 0; g2[3] = 0;
  i32x4 g3; g3[0] = 0; g3[1] = 0; g3[2] = 0; g3[3] = 0;
  i32x8 g4; g4[0] = 0; g4[1] = 0; g4[2] = 0; g4[3] = 0;
  g4[4] = 0; g4[5] = 0; g4[6] = 0; g4[7] = 0;
  __builtin_amdgcn_tensor_load_to_lds(g0, g1, g2, g3, g4, 0);
}
#endif

// W chunk = 16 pair-rows x 1024 floats -> LDS pair-rows of PSTRIDE.
__device__ __forceinline__ void issue_w(const float* Wt, int c, float* dst) {
#if HAS_TDM
  tdm_load((unsigned int)(uintptr_t)dst,
           (unsigned long long)(uintptr_t)(Wt + (size_t)c * 16 * 1024),
           1024u, 16u, 1u, 1024u, 16u, 0u, 1024ull, 0ull, 7u);
#endif
}

// Fallback cooperative copy (same padded layout) if TDM builtin is absent.
__device__ __forceinline__ void stage_w(const float* __restrict__ Wt, int c,
                                        float* __restrict__ dst, int tid) {
  for (int idx = tid * 4; idx < 16 * 1024; idx += 1024) {
    const int pr = idx >> 10, cc = idx & 1023;
    const float4 v =
        *(const float4*)(Wt + ((size_t)(c * 16 + pr)) * 1024 + cc);
    *(float4*)(dst + pr * PSTRIDE + cc + 8 * (cc >> 8)) = v;
  }
}

// ---------------------------------------------------------------------------
// One K-panel of WMMA work. ds_load offsets are compile-time immediates
// (<64KB) off per-chunk base pointers; B fragments arrive as adjacent VGPR
// pairs via single ds_load_b64 (k-pair interleaved layout).
// acc tile (mt,i): rows mt*16+[0,16), cols wave*64+i*16+[0,16).
// ---------------------------------------------------------------------------
template <int MTILES>
__device__ __forceinline__ void compute_chunk(
    const float* __restrict__ xs, const float* __restrict__ wbuf, int kc,
    v8f (&acc)[MTILES][4], int wave, int g, int ln) {
  const int cpad = (kc >= 256) ? 2 : 0;          // xs mid-row TDM pad
  const float* abase[MTILES];
#pragma unroll
  for (int mt = 0; mt < MTILES; ++mt)
    abase[mt] = xs + (mt * 16 + ln) * XS_STRIDE + kc + cpad + 2 * g;
  // B: pair-row p = 2t + g holds {W[k][f], W[k+1][f]} at float index
  //    p*PSTRIDE + f*2 + 8*(f>>7)  (intra-row TDM pads every 256 dw)
  const float* bfrag[4];
#pragma unroll
  for (int i = 0; i < 4; ++i) {
    const int fi = wave * 64 + i * 16;
    bfrag[i] = wbuf + g * PSTRIDE + fi * 2 + 8 * (fi >> 7) + ln * 2;
  }

#pragma unroll
  for (int t = 0; t < KC / 4; ++t) {             // 8 steps, fully unrolled
    v2f a[MTILES];
#pragma unroll
    for (int mt = 0; mt < MTILES; ++mt)
      a[mt] = *(const v2f*)(abase[mt] + t * 4);         // ds_load_b64
    v2f b[4];
#pragma unroll
    for (int i = 0; i < 4; ++i)
      b[i] = *(const v2f*)(bfrag[i] + t * 2 * PSTRIDE); // ds_load_b64
#pragma unroll
    for (int i = 0; i < 4; ++i)
#pragma unroll
      for (int mt = 0; mt < MTILES; ++mt)
        acc[mt][i] = __builtin_amdgcn_wmma_f32_16x16x4_f32(
            false, a[mt], false, b[i], (short)0, acc[mt][i], false, false);
  }
}

// Full K=512 sweep: 16 chunks of 32, TDM double-buffered + pipelined.
// Buffer selection is pure pointer arithmetic off wb0 so the LDS address
// space stays provable (a pointer array degrades loads to flat_load).
template <int MTILES>
__device__ __forceinline__ void run_gemm(
    const float* __restrict__ Wt, const float* __restrict__ xs,
    float* __restrict__ wb0, v8f (&acc)[MTILES][4], int tid, int wave, int g,
    int ln) {
#if HAS_TDM
  if (wave == 0) issue_w(Wt, 0, wb0);            // prefetch chunk 0
#endif
  for (int c = 0; c < 16; ++c) {
    __syncthreads();                              // buf[(c+1)&1] drained
#if HAS_TDM
    if (wave == 0) {
      if (c + 1 < 16) {
        issue_w(Wt, c + 1, wb0 + ((c + 1) & 1) * WB_PANEL);
        __builtin_amdgcn_s_wait_tensorcnt(1);     // chunk c landed (in-order)
      } else {
        __builtin_amdgcn_s_wait_tensorcnt(0);
      }
    }
#else
    stage_w(Wt, c, wb0 + (c & 1) * WB_PANEL, tid);
#endif
    __syncthreads();                              // buf[c&1] ready for all
    compute_chunk<MTILES>(xs, wb0 + (c & 1) * WB_PANEL, c * KC, acc, wave, g,
                          ln);
  }
}

// x tile (rows x 512) -> LDS rows of 516 (pad 2dw @256dw).
__device__ __forceinline__ void load_x_tile(const float* __restrict__ x,
                                            float* __restrict__ xs, int n,
                                            int s0, int rows, int tid,
                                            int wave) {
#if HAS_TDM
  if (wave == 0) {
    tdm_load((unsigned int)(uintptr_t)xs,
             (unsigned long long)(uintptr_t)(x + ((size_t)(n * S_LEN + s0)) * D_IN),
             512u, (unsigned)rows, 1u, 512u, (unsigned)rows, 0u,
             512ull, 0ull, 1u);
  }
#else
  for (int idx = tid * 2; idx < rows * 512; idx += 512) {
    const int r = idx >> 9, c = idx & 511;
    *(float2*)(xs + r * XS_STRIDE + c + (c >= 256 ? 2 : 0)) =
        *(const float2*)(x + ((size_t)(n * S_LEN + s0 + r)) * D_IN + c);
  }
#endif
}

// ---------------------------------------------------------------------------
// Pass 0: one-time weight interleave.  Wt[w][p][f*2+e] = W_w[k=2p+e][f],
// head-major f for Wq/Wk/Wv ([H][D][HD]), flat for Wo.  grid (256,4) x 256.
// ---------------------------------------------------------------------------
__global__ __launch_bounds__(256) void la_interleave(
    const float* __restrict__ Wq, const float* __restrict__ Wk,
    const float* __restrict__ Wv, const float* __restrict__ Wo,
    float* __restrict__ Wt) {
  const int w = blockIdx.y;                       // 0=q 1=k 2=v 3=o
  const int p = blockIdx.x;                       // k-pair 0..255
  const int f = threadIdx.x * 2;                  // 2 columns per thread
  const float* W = (w == 0) ? Wq : (w == 1) ? Wk : (w == 2) ? Wv : Wo;
  float e00, e01, e10, e11;                       // e{k-lsb}{f-off}
  if (w < 3) {
    const float* base = W + (size_t)(f >> 6) * 32768 + (f & 63);
    e00 = base[(size_t)(2 * p) * 64];
    e01 = base[(size_t)(2 * p) * 64 + 1];
    e10 = base[(size_t)(2 * p + 1) * 64];
    e11 = base[(size_t)(2 * p + 1) * 64 + 1];
  } else {
    const float* base = W + (size_t)(2 * p) * 512 + f;
    e00 = base[0];
    e01 = base[1];
    e10 = base[512];
    e11 = base[513];
  }
  float4 o;
  o.x = e00; o.y = e10; o.z = e01; o.w = e11;     // pairs adjacent
  *(float4*)(Wt + ((size_t)w * 256 + p) * 1024 + f * 2) = o;
}

// ---------------------------------------------------------------------------
// Pass 1: per 32-row tile compute K,V (register accumulators), reduce rows
// -> per-block partial KV / Ksum.  grid = (S/32, N), block = 256.
// ---------------------------------------------------------------------------
__global__ __launch_bounds__(256) void la_phase1(
    const float* __restrict__ x, const float* __restrict__ Wtk,
    const float* __restrict__ bk, const float* __restrict__ Wtv,
    const float* __restrict__ bv, float* __restrict__ pKV,
    float* __restrict__ pKS) {
  extern __shared__ float smem[];
  float* xs  = smem;                               // 32 x 516
  float* wb0 = smem + 32 * XS_STRIDE;              // 2 x WB_PANEL

  const int n = blockIdx.y, blk = blockIdx.x;
  const int s0 = blk * 32;
  const int tid = threadIdx.x, lane = tid & 31, wave = tid >> 5;
  const int g = lane >> 4, ln = lane & 15;

  load_x_tile(x, xs, n, s0, 32, tid, wave);

  v8f kacc[2][4], vacc[2][4];
  const v8f z = {};
#pragma unroll
  for (int mt = 0; mt < 2; ++mt)
#pragma unroll
    for (int i = 0; i < 4; ++i) { kacc[mt][i] = z; vacc[mt][i] = z; }

  run_gemm<2>(Wtk, xs, wb0, kacc, tid, wave, g, ln);
  run_gemm<2>(Wtv, xs, wb0, vacc, tid, wave, g, ln);

#pragma unroll
  for (int i = 0; i < 4; ++i) {
    const int f = wave * 64 + i * 16 + ln;
    const float bkf = bk[f], bvf = bv[f];
    float kv = 0.f, ks = 0.f;
#pragma unroll
    for (int mt = 0; mt < 2; ++mt)
#pragma unroll
      for (int j = 0; j < 8; ++j) {
        const float kvl = fmapf(kacc[mt][i][j] + bkf);
        const float vvl = vacc[mt][i][j] + bvf;
        kv += kvl * vvl;
        ks += kvl;
      }
    kv += __shfl_xor(kv, 16, 32);                 // merge the two row-halves
    ks += __shfl_xor(ks, 16, 32);
    if (lane < 16) {
      const size_t base = ((size_t)(n * 256 + blk)) * F_DIM + f;
      pKV[base] = kv;
      pKS[base] = ks;
    }
  }
}

// ---------------------------------------------------------------------------
// Pass 2: deterministic serial reduction of 256 block partials.
// ---------------------------------------------------------------------------
__global__ __launch_bounds__(512) void la_reduce(
    const float* __restrict__ pKV, const float* __restrict__ pKS,
    float* __restrict__ KVf, float* __restrict__ KSf) {
  const int n = blockIdx.x, f = threadIdx.x;
  float skv = 0.f, sks = 0.f;
  for (int b = 0; b < 256; ++b) {
    const size_t idx = ((size_t)(n * 256 + b)) * F_DIM + f;
    skv += pKV[idx];
    sks += pKS[idx];
  }
  KVf[n * F_DIM + f] = skv;
  KSf[n * F_DIM + f] = sks;
}

// ---------------------------------------------------------------------------
// Pass 3: GEMM Q, O = Q*KV/(Q*Ksum+eps) on fragments, overwrite x-tile LDS
// with O, GEMM with Wo, gelu, store.  grid = (S/64, N), block = 256.
// ---------------------------------------------------------------------------
__global__ __launch_bounds__(256) void la_phase3(
    const float* __restrict__ x, const float* __restrict__ Wtq,
    const float* __restrict__ bq, const float* __restrict__ Wto,
    const float* __restrict__ bo, const float* __restrict__ KVf,
    const float* __restrict__ KSf, float* __restrict__ out) {
  extern __shared__ float smem[];
  float* xs  = smem;                               // 64 x 516 (x, then O)
  float* wb0 = smem + 64 * XS_STRIDE;              // 2 x WB_PANEL

  const int n = blockIdx.y, blk = blockIdx.x;
  const int s0 = blk * 64;
  const int tid = threadIdx.x, lane = tid & 31, wave = tid >> 5;
  const int g = lane >> 4, ln = lane & 15;

  load_x_tile(x, xs, n, s0, 64, tid, wave);

  v8f acc[4][4];
  const v8f z = {};
#pragma unroll
  for (int mt = 0; mt < 4; ++mt)
#pragma unroll
    for (int i = 0; i < 4; ++i) acc[mt][i] = z;

  run_gemm<4>(Wtq, xs, wb0, acc, tid, wave, g, ln);

  __syncthreads();  // everyone done reading x -> overwrite with O
#pragma unroll
  for (int i = 0; i < 4; ++i) {
    const int f = wave * 64 + i * 16 + ln;
    const int fp = f + (f >= 256 ? 2 : 0);
    const float bqf = bq[f];
    const float kvf = KVf[n * F_DIM + f];
    const float ksf = KSf[n * F_DIM + f];
#pragma unroll
    for (int mt = 0; mt < 4; ++mt)
#pragma unroll
      for (int j = 0; j < 8; ++j) {
        const float q = fmapf(acc[mt][i][j] + bqf);
        const float o = q * kvf / (q * ksf + EPS);
        xs[(mt * 16 + j + 8 * g) * XS_STRIDE + fp] = o;
      }
  }
  __syncthreads();

#pragma unroll
  for (int mt = 0; mt < 4; ++mt)
#pragma unroll
    for (int i = 0; i < 4; ++i) acc[mt][i] = z;

  run_gemm<4>(Wto, xs, wb0, acc, tid, wave, g, ln);

#pragma unroll
  for (int i = 0; i < 4; ++i) {
    const int f = wave * 64 + i * 16 + ln;
    const float bof = bo[f];
#pragma unroll
    for (int mt = 0; mt < 4; ++mt)
#pragma unroll
      for (int j = 0; j < 8; ++j) {
        const int row = mt * 16 + j + 8 * g;
        out[((size_t)(n * S_LEN + s0 + row)) * OUT_DIM + f] =
            geluf(acc[mt][i][j] + bof);
      }
  }
}

// ---------------------------------------------------------------------------
extern "C" void kernel_launch(void* const* d_in, const int* in_sizes, int n_in,
                              void* d_out, int out_size, void* d_ws,
                              size_t ws_size, hipStream_t stream) {
  const float* x  = (const float*)d_in[0];
  const float* Wq = (const float*)d_in[1];
  const float* bq = (const float*)d_in[2];
  const float* Wk = (const float*)d_in[3];
  const float* bk = (const float*)d_in[4];
  const float* Wv = (const float*)d_in[5];
  const float* bv = (const float*)d_in[6];
  const float* Wo = (const float*)d_in[7];
  const float* bo = (const float*)d_in[8];
  float* out = (float*)d_out;

  float* ws  = (float*)d_ws;
  float* pKV = ws;                                 // [N][256][512]
  float* pKS = ws + (size_t)4 * 256 * 512;         // [N][256][512]
  float* KVf = ws + (size_t)2 * 4 * 256 * 512;     // [N][512]
  float* KSf = KVf + 4 * 512;                      // [N][512]
  float* Wt  = KSf + 4 * 512;                      // [4][256][1024] interleaved

  const size_t lds1 = (size_t)(32 * XS_STRIDE + 2 * WB_PANEL) * 4;  // ~201KB
  const size_t lds3 = (size_t)(64 * XS_STRIDE + 2 * WB_PANEL) * 4;  // ~267KB

  la_interleave<<<dim3(256, 4), 256, 0, stream>>>(Wq, Wk, Wv, Wo, Wt);
  la_phase1<<<dim3(S_LEN / 32, 4), 256, lds1, stream>>>(
      x, Wt + (size_t)1 * WT_SZ, bk, Wt + (size_t)2 * WT_SZ, bv, pKV, pKS);
  la_reduce<<<dim3(4), 512, 0, stream>>>(pKV, pKS, KVf, KSf);
  la_phase3<<<dim3(S_LEN / 64, 4), 256, lds3, stream>>>(
      x, Wt, bq, Wt + (size_t)3 * WT_SZ, bo, KVf, KSf, out);
}